// Rwkv5SelfAttention_9337258902330
// MI455X (gfx1250) — compile-verified
//
#include <hip/hip_runtime.h>

typedef float v2f __attribute__((ext_vector_type(2)));
typedef float v4f __attribute__((ext_vector_type(4)));
typedef float v8f __attribute__((ext_vector_type(8)));

#define DD  4096
#define HH  64
#define SS  64
#define EPS 1e-5f

// ---------------------------------------------------------------------------
// Kernel 1: LayerNorm(x) -> xx, state1_out, and the four time-mix vectors.
// One 1024-thread block (32 waves), 4 elements/thread.
// ---------------------------------------------------------------------------
__global__ __launch_bounds__(1024) void ln_mix_kernel(
    const float* __restrict__ x, const float* __restrict__ state1,
    const float* __restrict__ tmk, const float* __restrict__ tmv,
    const float* __restrict__ tmr, const float* __restrict__ tmg,
    const float* __restrict__ ln1_w, const float* __restrict__ ln1_b,
    float* __restrict__ xx_out, float* __restrict__ mixes,
    float* __restrict__ state1_out)
{
    __shared__ float s_sum[32], s_sq[32];
    const int tid = threadIdx.x;
    const int wave = tid >> 5, lane = tid & 31;

    float vals[4];
    float lsum = 0.f, lsq = 0.f;
    #pragma unroll
    for (int i = 0; i < 4; ++i) {
        float a = x[tid + i * 1024];
        vals[i] = a; lsum += a; lsq += a * a;
    }
    #pragma unroll
    for (int off = 16; off; off >>= 1) {
        lsum += __shfl_xor(lsum, off, 32);
        lsq  += __shfl_xor(lsq,  off, 32);
    }
    if (lane == 0) { s_sum[wave] = lsum; s_sq[wave] = lsq; }
    __syncthreads();
    if (wave == 0) {
        float a = s_sum[lane], b = s_sq[lane];
        #pragma unroll
        for (int off = 16; off; off >>= 1) {
            a += __shfl_xor(a, off, 32);
            b += __shfl_xor(b, off, 32);
        }
        if (lane == 0) { s_sum[0] = a; s_sq[0] = b; }
    }
    __syncthreads();
    const float mean = s_sum[0] * (1.0f / DD);
    const float var  = s_sq[0] * (1.0f / DD) - mean * mean;
    const float rstd = rsqrtf(var + EPS);

    #pragma unroll
    for (int i = 0; i < 4; ++i) {
        int idx = tid + i * 1024;
        float xxv = (vals[i] - mean) * rstd * ln1_w[idx] + ln1_b[idx];
        xx_out[idx]     = xxv;
        state1_out[idx] = xxv;
        float s1 = state1[idx];
        mixes[0 * DD + idx] = s1 + (xxv - s1) * tmr[idx];  // for Wr
        mixes[1 * DD + idx] = s1 + (xxv - s1) * tmk[idx];  // for Wk
        mixes[2 * DD + idx] = s1 + (xxv - s1) * tmv[idx];  // for Wv
        mixes[3 * DD + idx] = s1 + (xxv - s1) * tmg[idx];  // for Wg
    }
}

// ---------------------------------------------------------------------------
// Kernel 2: fused GEMV for Wr/Wk/Wv/Wg (16384 rows total). One wave per row,
// coalesced float4 nontemporal weight streaming, shuffle reduction.
// This is the bandwidth-critical path (268 MB one-shot).
// ---------------------------------------------------------------------------
__global__ __launch_bounds__(256) void gemv4_kernel(
    const float* __restrict__ Wr, const float* __restrict__ Wk,
    const float* __restrict__ Wv, const float* __restrict__ Wg,
    const float* __restrict__ mixes, float* __restrict__ out)
{
    const int wave = threadIdx.x >> 5, lane = threadIdx.x & 31;
    const int row  = blockIdx.x * 8 + wave;         // 0..16383
    const int mat  = row >> 12;                     // 0..3
    const int r12  = row & 4095;
    const float* W = (mat == 0) ? Wr : (mat == 1) ? Wk : (mat == 2) ? Wv : Wg;
    const v4f* Wrow = (const v4f*)(W + (size_t)r12 * DD);
    const v4f* m4   = (const v4f*)(mixes + mat * DD);

    float acc = 0.f;
    #pragma unroll 4
    for (int j = lane; j < DD / 4; j += 32) {
        v4f a = __builtin_nontemporal_load(&Wrow[j]);  // stream weights, TH=NT
        v4f b = m4[j];                                 // mix vector: L2-resident
        acc = fmaf(a.x, b.x, acc); acc = fmaf(a.y, b.y, acc);
        acc = fmaf(a.z, b.z, acc); acc = fmaf(a.w, b.w, acc);
    }
    #pragma unroll
    for (int off = 16; off; off >>= 1) acc += __shfl_xor(acc, off, 32);
    if (lane == 0) out[row] = acc;
}

// ---------------------------------------------------------------------------
// Kernel 3: per-head attention state math. One wave32 per head.
//   wkv[n] = (r.(k*tf)) * v[n] + (r @ state2)[n]      <- WMMA f32 16x16x4
//   state2_out = k v^T + state2 * td
//   instance-norm(wkv) * lnx -> gated by SiLU(g) -> y
// A-matrix = broadcast r chunk (known ISA layout: lanes 0-15 K=0/1 in
// VGPR0/1, lanes 16-31 K=2/3); B-matrix = 4x16 state2 tile (row striped
// across lanes). Rows of D replicate the 1xN result; lane n holds rs2[n0+n].
// ---------------------------------------------------------------------------
__global__ __launch_bounds__(32) void head_kernel(
    const float* __restrict__ rkvg,     // [r | k | v | g_lin], each DD
    const float* __restrict__ state2,   // H*S*S
    const float* __restrict__ td, const float* __restrict__ tf,
    const float* __restrict__ lnx_w, const float* __restrict__ lnx_b,
    float* __restrict__ y, float* __restrict__ state2_out)
{
    const int h = blockIdx.x, lane = threadIdx.x;
    const float* rh  = rkvg + h * SS;
    const float* kh  = rkvg + DD + h * SS;
    const float* vh  = rkvg + 2 * DD + h * SS;
    const float* gh  = rkvg + 3 * DD + h * SS;
    const float* S2  = state2 + (size_t)h * SS * SS;
    float*       SO  = state2_out + (size_t)h * SS * SS;
    const float* tdh = td + h * SS;
    const float* tfh = tf + h * SS;

    // tfdot = sum_s r[s]*k[s]*tf[s]
    float t0 = rh[lane] * kh[lane] * tfh[lane]
             + rh[lane + 32] * kh[lane + 32] * tfh[lane + 32];
    #pragma unroll
    for (int off = 16; off; off >>= 1) t0 += __shfl_xor(t0, off, 32);
    const float tfdot = t0;

    // rs2[n] = sum_s r[s] * state2[s][n] via V_WMMA_F32_16X16X4_F32
    const int half = lane >> 4, l15 = lane & 15;
    const int ka = half ? 2 : 0, kb = half ? 3 : 1;
    float rs2c[4];
    #pragma unroll
    for (int c = 0; c < 4; ++c) {
        const int n0 = c * 16;
        v8f acc = {};
        for (int s0 = 0; s0 < SS; s0 += 4) {
            v2f A, B;
            A.x = rh[s0 + ka];                      // A[m][k] = r[s0+k] (all m)
            A.y = rh[s0 + kb];
            B.x = S2[(s0 + ka) * SS + n0 + l15];    // B[k][n] = state2[s0+k][n0+n]
            B.y = S2[(s0 + kb) * SS + n0 + l15];
            acc = __builtin_amdgcn_wmma_f32_16x16x4_f32(
                false, A, false, B, (short)0, acc, false, false);
        }
        rs2c[c] = acc[0];   // rows replicated; lane l15 holds rs2[n0+l15]
    }
    const float rs2_a = (lane < 16) ? rs2c[0] : rs2c[1];  // n = lane
    const float rs2_b = (lane < 16) ? rs2c[2] : rs2c[3];  // n = lane + 32

    // state2_out = k v^T + state2 * td   (coalesced streaming update)
    for (int idx = lane; idx < SS * SS; idx += 32) {
        int s = idx >> 6, n = idx & 63;
        SO[idx] = kh[s] * vh[n] + S2[idx] * tdh[s];
    }

    // wkv + per-head instance norm + SiLU gate
    const int na = lane, nb = lane + 32;
    float wa = tfdot * vh[na] + rs2_a;
    float wb = tfdot * vh[nb] + rs2_b;
    float sum = wa + wb, sq = wa * wa + wb * wb;
    #pragma unroll
    for (int off = 16; off; off >>= 1) {
        sum += __shfl_xor(sum, off, 32);
        sq  += __shfl_xor(sq,  off, 32);
    }
    const float mu   = sum * (1.0f / SS);
    const float var  = sq * (1.0f / SS) - mu * mu;
    const float rstd = rsqrtf(var + EPS);
    const int ga = h * SS + na, gb = h * SS + nb;
    float ya = (wa - mu) * rstd * lnx_w[ga] + lnx_b[ga];
    float yb = (wb - mu) * rstd * lnx_w[gb] + lnx_b[gb];
    float g1 = gh[na], g2 = gh[nb];
    ya *= g1 / (1.0f + __expf(-g1));   // SiLU
    yb *= g2 / (1.0f + __expf(-g2));
    y[ga] = ya; y[gb] = yb;
}

// ---------------------------------------------------------------------------
// Kernel 4: out = x + Wo @ y   (4096 rows, wave per row, NT weight stream)
// ---------------------------------------------------------------------------
__global__ __launch_bounds__(256) void gemv_out_kernel(
    const float* __restrict__ Wo, const float* __restrict__ y,
    const float* __restrict__ x, float* __restrict__ out)
{
    const int wave = threadIdx.x >> 5, lane = threadIdx.x & 31;
    const int row = blockIdx.x * 8 + wave;
    const v4f* Wrow = (const v4f*)(Wo + (size_t)row * DD);
    const v4f* y4   = (const v4f*)y;

    float acc = 0.f;
    #pragma unroll 4
    for (int j = lane; j < DD / 4; j += 32) {
        v4f a = __builtin_nontemporal_load(&Wrow[j]);
        v4f b = y4[j];
        acc = fmaf(a.x, b.x, acc); acc = fmaf(a.y, b.y, acc);
        acc = fmaf(a.z, b.z, acc); acc = fmaf(a.w, b.w, acc);
    }
    #pragma unroll
    for (int off = 16; off; off >>= 1) acc += __shfl_xor(acc, off, 32);
    if (lane == 0) out[row] = x[row] + acc;
}

// ---------------------------------------------------------------------------
extern "C" void kernel_launch(void* const* d_in, const int* in_sizes, int n_in,
                              void* d_out, int out_size, void* d_ws, size_t ws_size,
                              hipStream_t stream) {
    const float* x      = (const float*)d_in[0];
    const float* state1 = (const float*)d_in[1];
    const float* state2 = (const float*)d_in[2];
    const float* tmk    = (const float*)d_in[3];
    const float* tmv    = (const float*)d_in[4];
    const float* tmr    = (const float*)d_in[5];
    const float* tmg    = (const float*)d_in[6];
    const float* td     = (const float*)d_in[7];   // (H,S,1) flat
    const float* tf     = (const float*)d_in[8];   // (H,S,1) flat
    const float* Wr     = (const float*)d_in[9];
    const float* Wk     = (const float*)d_in[10];
    const float* Wv     = (const float*)d_in[11];
    const float* Wg     = (const float*)d_in[12];
    const float* Wo     = (const float*)d_in[13];
    const float* ln1_w  = (const float*)d_in[14];
    const float* ln1_b  = (const float*)d_in[15];
    const float* lnx_w  = (const float*)d_in[16];
    const float* lnx_b  = (const float*)d_in[17];

    float* ws    = (float*)d_ws;
    float* xx    = ws;              // D
    float* mixes = ws + DD;         // 4*D  (order: r,k,v,g mixes)
    float* rkvg  = ws + 5 * DD;     // 4*D  (r, k, v, g_lin)
    float* yv    = ws + 9 * DD;     // D

    float* out = (float*)d_out;          // D
    float* s1o = out + DD;               // D
    float* s2o = out + 2 * DD;           // H*S*S

    ln_mix_kernel<<<1, 1024, 0, stream>>>(x, state1, tmk, tmv, tmr, tmg,
                                          ln1_w, ln1_b, xx, mixes, s1o);
    gemv4_kernel<<<2048, 256, 0, stream>>>(Wr, Wk, Wv, Wg, mixes, rkvg);
    head_kernel<<<HH, 32, 0, stream>>>(rkvg, state2, td, tf, lnx_w, lnx_b,
                                       yv, s2o);
    gemv_out_kernel<<<512, 256, 0, stream>>>(Wo, yv, x, out);
}